// VectorQuantizerEMA_5274219840274
// MI455X (gfx1250) — compile-verified
//
#include <hip/hip_runtime.h>
#include <hip/hip_bf16.h>
#include <math.h>

// ---------------------------------------------------------------------------
// VectorQuantizerEMA eval forward for MI455X (gfx1250, wave32, WMMA + TDM).
//
// inputs (B=32, C=D=64, H=64, W=64) f32, embedding (K=512, D=64) f32.
// N = 131072 rows. d_out (flat, return order):
//   [0] loss | [1..8388608] quantized_st (B,C,H,W) | [8388609] perplexity |
//   [8388610..] indices (as float)
// ---------------------------------------------------------------------------

#define DIM       64
#define KCODES    512
#define NROWS     131072
#define HWSZ      4096
#define CHW       262144
#define OUT_ELE   8388608
#define OUT_OFF   1
#define PERP_OFF  (1 + OUT_ELE)
#define IDX_OFF   (2 + OUT_ELE)

#define EMB_STRIDE 68                    // 64 data + 4 pad dwords (bank-free)
#define NBLK       512                   // 256 rows per block
#define SMEM_FLOATS (KCODES * EMB_STRIDE + KCODES)   // emb + norms
#define SMEM_BYTES  (SMEM_FLOATS * 4 + 256 * 4 + 8 * 4)

typedef __attribute__((ext_vector_type(2))) float v2f;
typedef __attribute__((ext_vector_type(8))) float v8f;
typedef __attribute__((ext_vector_type(4))) unsigned int u32x4;
typedef __attribute__((ext_vector_type(8))) int i32x8;
typedef __attribute__((ext_vector_type(4))) int i32x4;

// --------------------------- kernel 1: prep --------------------------------
__global__ void vq_prep(const float* __restrict__ emb,
                        float* __restrict__ norms,
                        unsigned int* __restrict__ counts) {
    int k = threadIdx.x;                       // 512 threads, 1 block
    const float4* e4 = (const float4*)(emb + k * DIM);
    float s = 0.f;
#pragma unroll
    for (int i = 0; i < DIM / 4; ++i) {
        float4 v = e4[i];
        s += v.x * v.x + v.y * v.y + v.z * v.z + v.w * v.w;
    }
    norms[k] = s;
    counts[k] = 0u;                            // must re-zero every call
}

// ------------- TDM: DMA the 512x64 f32 codebook into padded LDS ------------
// D# per ISA §8: 2D tensor, data_size=4B, tile=tensor=512x64, pad_enable so
// each 64-dword row is followed by 4 pad dwords (LDS row stride = 68).
__device__ __forceinline__ void tdm_load_codebook(unsigned lds_byte_addr,
                                                  const float* emb) {
    unsigned long long ga = (unsigned long long)(uintptr_t)emb;
    u32x4 g0 = {
        1u,                                           // count=1, user mode
        lds_byte_addr,                                // lds_addr
        (unsigned)(ga & 0xFFFFFFFFu),                 // global_addr[31:0]
        (unsigned)((ga >> 32) & 0x01FFFFFFu) | (2u << 30) // addr[56:32]|type=2
    };
    unsigned d0 = (2u << 16)        // data_size = 4B
                | (1u << 20)        // pad_enable
                | (5u << 22)        // pad_interval: 2^(5+1)=64 dwords
                | (3u << 25);       // pad_amount: 3+1 = 4 dwords
    i32x8 g1 = {
        (int)d0,
        (int)((DIM & 0xFFFFu) << 16),                 // tensor_dim0 lo16
        (int)(((DIM >> 16) & 0xFFFFu) | ((KCODES & 0xFFFFu) << 16)),
        (int)(((KCODES >> 16) & 0xFFFFu) | (DIM << 16)),  // tile_dim0 = 64
        (int)(KCODES),                                // tile_dim1 = 512
        (int)(DIM),                                   // tensor_dim0_stride
        0, 0
    };
    i32x4 g2 = {0, 0, 0, 0};                          // 2D: groups 2/3 unused
    i32x4 g3 = {0, 0, 0, 0};
    i32x8 g4 = {0, 0, 0, 0, 0, 0, 0, 0};
    // This toolchain declares the 6-arg form (probe: "expected 6").
    __builtin_amdgcn_tensor_load_to_lds(g0, g1, g2, g3, g4, 0);
}

// --------------------------- kernel 2: main --------------------------------
// 8 waves/block, 2 row-tiles (32 rows) per wave; B fragments from LDS shared
// by two WMMA accumulator chains.
__global__ void __launch_bounds__(256)
vq_main(const float* __restrict__ inp,     // B,C,H,W
        const float* __restrict__ emb,     // K,D
        const float* __restrict__ norms,   // K
        unsigned int* __restrict__ counts, // K
        float* __restrict__ partials,      // one SSE partial per block
        float* __restrict__ out) {
    extern __shared__ float smem[];
    float* sEmb  = smem;                                // 512*68
    float* sNorm = smem + KCODES * EMB_STRIDE;          // 512
    int*   sIdx  = (int*)(sNorm + KCODES);              // 256
    float* sWsum = (float*)(sIdx + 256);                // 8

    const int tid  = threadIdx.x;
    const int w    = tid >> 5;
    const int lane = tid & 31;
    const int m16  = lane & 15;
    const int hi   = lane >> 4;

    // stage norms (trivial) + codebook via TDM (wave 0 issues; TENSORcnt)
    sNorm[tid]       = norms[tid];
    sNorm[tid + 256] = norms[tid + 256];
    if (w == 0) {
        tdm_load_codebook((unsigned)(uintptr_t)sEmb, emb);
        __builtin_amdgcn_s_wait_tensorcnt(0);
    }
    __syncthreads();

    const int n0 = (blockIdx.x * 8 + w) * 32;           // 32 rows per wave

    // ---- A fragments, two 16-row tiles (ISA 16x4 f32 layout) ----
    v2f a0[16], a1[16];
    {
        const int nA = n0 + m16;
        const float* xb0 = inp + (size_t)(nA >> 12) * CHW + (nA & 4095);
        const int nB = n0 + 16 + m16;
        const float* xb1 = inp + (size_t)(nB >> 12) * CHW + (nB & 4095);
#pragma unroll
        for (int s = 0; s < 16; ++s) {
            const int d0 = 4 * s + 2 * hi;
            a0[s].x = xb0[(size_t)d0 * HWSZ];
            a0[s].y = xb0[(size_t)(d0 + 1) * HWSZ];
            a1[s].x = xb1[(size_t)d0 * HWSZ];
            a1[s].y = xb1[(size_t)(d0 + 1) * HWSZ];
        }
    }

    float bv0[8], bv1[8];
    int   bi0[8], bi1[8];
#pragma unroll
    for (int r = 0; r < 8; ++r) {
        bv0[r] = 3.4e38f; bi0[r] = 0;
        bv1[r] = 3.4e38f; bi1[r] = 0;
    }

    for (int c = 0; c < KCODES / 16; ++c) {
        const int kcol = c * 16 + m16;
        const float* eb = sEmb + kcol * EMB_STRIDE + 2 * hi;
        v8f acc0 = {}, acc1 = {};
#pragma unroll
        for (int s = 0; s < 16; ++s) {
            const v2f bf = *reinterpret_cast<const v2f*>(eb + 4 * s);
            acc0 = __builtin_amdgcn_wmma_f32_16x16x4_f32(
                false, a0[s], false, bf, (short)0, acc0, false, false);
            acc1 = __builtin_amdgcn_wmma_f32_16x16x4_f32(
                false, a1[s], false, bf, (short)0, acc1, false, false);
        }
        const float en = sNorm[kcol];
#pragma unroll
        for (int r = 0; r < 8; ++r) {
            const float s0 = fmaf(-2.0f, acc0[r], en);
            if (s0 < bv0[r] || (s0 == bv0[r] && kcol < bi0[r])) {
                bv0[r] = s0; bi0[r] = kcol;
            }
            const float s1 = fmaf(-2.0f, acc1[r], en);
            if (s1 < bv1[r] || (s1 == bv1[r] && kcol < bi1[r])) {
                bv1[r] = s1; bi1[r] = kcol;
            }
        }
    }

    // ---- argmin reduce across the 16-lane N group (ties -> lower index) ---
#pragma unroll
    for (int r = 0; r < 8; ++r) {
        float v0 = bv0[r]; int i0 = bi0[r];
        float v1 = bv1[r]; int i1 = bi1[r];
#pragma unroll
        for (int off = 8; off >= 1; off >>= 1) {
            float ov = __shfl_xor(v0, off, 32); int oi = __shfl_xor(i0, off, 32);
            if (ov < v0 || (ov == v0 && oi < i0)) { v0 = ov; i0 = oi; }
            ov = __shfl_xor(v1, off, 32); oi = __shfl_xor(i1, off, 32);
            if (ov < v1 || (ov == v1 && oi < i1)) { v1 = ov; i1 = oi; }
        }
        bi0[r] = i0; bi1[r] = i1;
    }

    if (m16 == 0) {                     // lanes 0 and 16: rows 0-7 / 8-15
        const int base = w * 32;
#pragma unroll
        for (int r = 0; r < 8; ++r) {
            const int m = r + 8 * hi;
            int k = bi0[r];
            sIdx[base + m] = k;
            out[IDX_OFF + n0 + m] = (float)k;
            atomicAdd(&counts[k], 1u);
            k = bi1[r];
            sIdx[base + 16 + m] = k;
            out[IDX_OFF + n0 + 16 + m] = (float)k;
            atomicAdd(&counts[k], 1u);
        }
    }
    __syncthreads();

    // ---- phase 2: gather codes, write B,C,H,W output, accumulate SSE ------
    const int N0   = blockIdx.x * 256;
    const int bb   = N0 >> 12;
    const int rem0 = N0 & 4095;
    float sse = 0.f;
#pragma unroll 4
    for (int i = 0; i < 64; ++i) {
        const int t  = tid + i * 256;       // 0..16383
        const int d  = t >> 8;              // 0..63
        const int ro = t & 255;             // row in block tile
        const int k  = sIdx[ro];
        const float e  = sEmb[k * EMB_STRIDE + d];   // from staged LDS copy
        const size_t off = (size_t)bb * CHW + (size_t)d * HWSZ + rem0 + ro;
        const float x  = inp[off];
        out[OUT_OFF + off] = e;             // straight-through == quantized
        const float df = e - x;
        sse = fmaf(df, df, sse);
    }
#pragma unroll
    for (int off = 16; off >= 1; off >>= 1) sse += __shfl_xor(sse, off, 32);
    if (lane == 0) sWsum[w] = sse;
    __syncthreads();
    if (tid == 0) {
        float s = 0.f;
#pragma unroll
        for (int j = 0; j < 8; ++j) s += sWsum[j];
        partials[blockIdx.x] = s;
    }
}

// --------------------------- kernel 3: finalize ----------------------------
__global__ void vq_fin(const unsigned int* __restrict__ counts,
                       const float* __restrict__ partials,
                       float* __restrict__ out) {
    __shared__ float red[512];
    const int k = threadIdx.x;
    const float p = (float)counts[k] * (1.0f / (float)NROWS);
    red[k] = p * logf(p + 1e-10f);
    __syncthreads();
#pragma unroll
    for (int s = 256; s >= 1; s >>= 1) {   // fixed-order tree: deterministic
        if (k < s) red[k] += red[k + s];
        __syncthreads();
    }
    if (k == 0) {
        out[PERP_OFF] = expf(-red[0]);
        float sse = 0.f;
        for (int j = 0; j < NBLK; ++j) sse += partials[j];
        out[0] = 0.25f * sse * (1.0f / (float)OUT_ELE);
    }
}

// ---------------------------------------------------------------------------
extern "C" void kernel_launch(void* const* d_in, const int* in_sizes, int n_in,
                              void* d_out, int out_size, void* d_ws, size_t ws_size,
                              hipStream_t stream) {
    const float* inp = (const float*)d_in[0];   // (32,64,64,64) f32
    const float* emb = (const float*)d_in[1];   // (512,64) f32
    float* out = (float*)d_out;

    float*        ws     = (float*)d_ws;
    float*        norms  = ws;                              // 512 f32
    unsigned int* counts = (unsigned int*)(ws + 512);       // 512 u32
    float*        parts  = ws + 1024;                       // 512 f32

    // allow ~142 KB dynamic LDS (host-side attr; not a stream op)
    (void)hipFuncSetAttribute((const void*)vq_main,
                              hipFuncAttributeMaxDynamicSharedMemorySize,
                              SMEM_BYTES);

    vq_prep<<<1,    512, 0,          stream>>>(emb, norms, counts);
    vq_main<<<NBLK, 256, SMEM_BYTES, stream>>>(inp, emb, norms, counts, parts, out);
    vq_fin <<<1,    512, 0,          stream>>>(counts, parts, out);
}